// MLFormer_4131758538810
// MI455X (gfx1250) — compile-verified
//
#include <hip/hip_runtime.h>
#include <hip/hip_bf16.h>

typedef __attribute__((ext_vector_type(2))) float v2f;
typedef __attribute__((ext_vector_type(8))) float v8f;

#define WMMA_F32X4(a, b, c) \
    __builtin_amdgcn_wmma_f32_16x16x4_f32(false, (a), false, (b), (short)0, (c), false, false)

// ---------------------------------------------------------------------------
// K0: dense rank-32 outer product: out[n, m] = sum_r U[n,r] * V[m,r]
// ---------------------------------------------------------------------------
__global__ void __launch_bounds__(256) k_rankmm(const float* __restrict__ U,
                                                const float* __restrict__ V,
                                                float* __restrict__ out,
                                                int rows, int cols) {
    int idx = blockIdx.x * 256 + threadIdx.x;
    if (idx >= rows * cols) return;
    int n = idx / cols;
    int m = idx - n * cols;
    const float* up = U + n * 32;
    const float* vp = V + m * 32;
    float acc = 0.f;
#pragma unroll
    for (int r = 0; r < 32; ++r) acc += up[r] * vp[r];
    out[idx] = acc;
}

// ---------------------------------------------------------------------------
// K1: WMMA projections. One wave handles a 16-row M-tile of the 512 rows.
//   t   = uH(16x1025)  @ KHH_V(1025x32)          (A strided by 8 in uh)
//   s   = uh(16x8193)  @ Kbd_U(8193x32)          -> sbuf
//   wh0 = (t*H*8)(16x32) @ KHH_U^T(32x1025)      -> wh0 buffer
// Main K-loops are guard-free (K-1 is a multiple of 4); the single remainder
// column is a branch-free epilogue using clamped addresses + A-operand zeroing.
// grid = 4 blocks x 256 threads (8 waves) = 32 M-tiles
// ---------------------------------------------------------------------------
__global__ void __launch_bounds__(256) k_wmma_proj(const float* __restrict__ uh,
                                                   const float* __restrict__ KHH_U,
                                                   const float* __restrict__ KHH_V,
                                                   const float* __restrict__ Kbd_U,
                                                   float* __restrict__ wh0,
                                                   float* __restrict__ sbuf) {
    __shared__ float tld[8][16 * 32];
    const int wave = threadIdx.x >> 5;
    const int lane = threadIdx.x & 31;
    const int half = lane >> 4;     // 0: lanes 0-15, 1: lanes 16-31
    const int mrow = lane & 15;
    const int wid  = blockIdx.x * 8 + wave;   // 0..31
    const int R    = wid * 16;                // row base into 512 rows
    const float HK = 1.0f / 1024.0f;          // H * 2^K = (1/8192)*8

    const float* arow = uh + (size_t)(R + mrow) * 8193;  // this lane's A row

    // ---- phase A: t tile (16x32), K = 1025, A column k maps to uh col 8k
    v8f t0 = {}; v8f t1 = {};
    for (int k0 = 0; k0 < 1024; k0 += 4) {            // guard-free main loop
        int ka = k0 + 2 * half;                        // <= 1022
        int kb = ka + 1;                               // <= 1023
        v2f a, b0, b1;
        a.x  = arow[8 * ka];
        a.y  = arow[8 * kb];
        b0.x = KHH_V[ka * 32 + mrow];
        b0.y = KHH_V[kb * 32 + mrow];
        b1.x = KHH_V[ka * 32 + 16 + mrow];
        b1.y = KHH_V[kb * 32 + 16 + mrow];
        t0 = WMMA_F32X4(a, b0, t0);
        t1 = WMMA_F32X4(a, b1, t1);
    }
    {   // epilogue: k = 1024 only (valid in half 0, slot x). Zero A elsewhere;
        // B loads use the clamped row 1024 (a[k]==0 kills any garbage product).
        float alast = arow[8192];
        v2f a, b0, b1;
        a.x  = (half == 0) ? alast : 0.f;
        a.y  = 0.f;
        b0.x = KHH_V[1024 * 32 + mrow];
        b0.y = b0.x;
        b1.x = KHH_V[1024 * 32 + 16 + mrow];
        b1.y = b1.x;
        t0 = WMMA_F32X4(a, b0, t0);
        t1 = WMMA_F32X4(a, b1, t1);
    }

    // ---- phase B: s tile (16x32), K = 8193, stride-1 A
    v8f s0 = {}; v8f s1 = {};
    for (int k0 = 0; k0 < 8192; k0 += 4) {            // guard-free main loop
        int ka = k0 + 2 * half;
        int kb = ka + 1;
        v2f a, b0, b1;
        a.x  = arow[ka];
        a.y  = arow[kb];
        b0.x = Kbd_U[ka * 32 + mrow];
        b0.y = Kbd_U[kb * 32 + mrow];
        b1.x = Kbd_U[ka * 32 + 16 + mrow];
        b1.y = Kbd_U[kb * 32 + 16 + mrow];
        s0 = WMMA_F32X4(a, b0, s0);
        s1 = WMMA_F32X4(a, b1, s1);
    }
    {   // epilogue: k = 8192 only
        float alast = arow[8192];
        v2f a, b0, b1;
        a.x  = (half == 0) ? alast : 0.f;
        a.y  = 0.f;
        b0.x = Kbd_U[8192 * 32 + mrow];
        b0.y = b0.x;
        b1.x = Kbd_U[8192 * 32 + 16 + mrow];
        b1.y = b1.x;
        s0 = WMMA_F32X4(a, b0, s0);
        s1 = WMMA_F32X4(a, b1, s1);
    }
    // store s tile: acc[j] = C[m = j + 8*half][n = mrow]
#pragma unroll
    for (int j = 0; j < 8; ++j) {
        int m = j + 8 * half;
        sbuf[(size_t)(R + m) * 32 + mrow]      = s0[j];
        sbuf[(size_t)(R + m) * 32 + 16 + mrow] = s1[j];
    }

    // stage t*HK into LDS row-major so it can be reloaded in A-operand layout
    float* tl = tld[wave];
#pragma unroll
    for (int j = 0; j < 8; ++j) {
        int m = j + 8 * half;
        tl[m * 32 + mrow]      = t0[j] * HK;
        tl[m * 32 + 16 + mrow] = t1[j] * HK;
    }
    __syncthreads();

    // ---- phase C: wh0 rows R..R+15 = tld(16x32) @ KHH_U^T, N = 1025
    // tiles 0..63 fully in range -> no guards
    for (int nt = 0; nt < 64; ++nt) {
        int n = nt * 16 + mrow;                       // <= 1023
        v8f acc = {};
#pragma unroll
        for (int k0 = 0; k0 < 32; k0 += 4) {
            int ka = k0 + 2 * half;
            int kb = ka + 1;
            v2f a, b;
            a.x = tl[mrow * 32 + ka];
            a.y = tl[mrow * 32 + kb];
            b.x = KHH_U[n * 32 + ka];
            b.y = KHH_U[n * 32 + kb];
            acc = WMMA_F32X4(a, b, acc);
        }
#pragma unroll
        for (int j = 0; j < 8; ++j) {
            int m = j + 8 * half;
            wh0[(size_t)(R + m) * 1025 + n] = acc[j];
        }
    }
    {   // last tile: only column n = 1024 is real; clamp B addresses, store col 0
        v8f acc = {};
#pragma unroll
        for (int k0 = 0; k0 < 32; k0 += 4) {
            int ka = k0 + 2 * half;
            int kb = ka + 1;
            v2f a, b;
            a.x = tl[mrow * 32 + ka];
            a.y = tl[mrow * 32 + kb];
            b.x = KHH_U[1024 * 32 + ka];   // same row for all lanes; cols n>1024
            b.y = KHH_U[1024 * 32 + kb];   // are garbage but never stored
            acc = WMMA_F32X4(a, b, acc);
        }
        if (mrow == 0) {
#pragma unroll
            for (int j = 0; j < 8; ++j) {
                int m = j + 8 * half;
                wh0[(size_t)(R + m) * 1025 + 1024] = acc[j];
            }
        }
    }
}

// ---------------------------------------------------------------------------
// K2: fused 3-level refinement, one block per (b,c) row.
// wh ping-pongs in static LDS (wA:4097 used max, wB:8193). uh gathers go
// through WGP$ (32KB row stays cache-resident; avoids LDS bank conflicts).
// ---------------------------------------------------------------------------
__global__ void __launch_bounds__(256) k_refine(const float* __restrict__ uh,
                                                const float* __restrict__ wh0,
                                                const float* __restrict__ sbuf,
                                                const float* __restrict__ Kbd_V,
                                                const float* __restrict__ Ke0,
                                                const float* __restrict__ Ko0,
                                                const float* __restrict__ Ke1,
                                                const float* __restrict__ Ko1,
                                                const float* __restrict__ Ke2,
                                                const float* __restrict__ Ko2,
                                                const int* __restrict__ nbrL0,  // nbr2 (2049x15)
                                                const int* __restrict__ nbrL1,  // nbr1 (4097x15)
                                                const int* __restrict__ nbrL2,  // nbr0 (8193x15)
                                                float* __restrict__ out) {
    __shared__ float wA[4100];
    __shared__ float wB[8200];
    const int tid = threadIdx.x;
    const int bc  = blockIdx.x;                 // 0..511
    const size_t base = (size_t)bc * 8193;
    const float* u = uh + base;                 // gathered via cache
    const float H = 1.0f / 8192.0f;

    for (int n = tid; n < 1025; n += 256) wA[n] = wh0[(size_t)bc * 1025 + n];
    __syncthreads();

    const float* KeP[3]  = {Ke0, Ke1, Ke2};
    const float* KoP[3]  = {Ko0, Ko1, Ko2};
    const int*   nbrP[3] = {nbrL0, nbrL1, nbrL2};
    float* oldW = wA;
    float* newW = wB;

#pragma unroll
    for (int i = 0; i < 3; ++i) {
        const int Nc = 1025 << i;           // 1025, 2049, 4097
        const int step = 4 >> i;            // 4, 2, 1 (compile-time after unroll)
        const float hf = H * (float)step;
        const float* Ke = KeP[i];
        const float* Ko = KoP[i];
        const int* nbr  = nbrP[i];
        // even outputs: new[2n] = old[n] + hf * sum_m Ke[n,m]*u[step*nbr[2n,2m]]
        for (int n = tid; n < Nc; n += 256) {
            const int* row = nbr + (size_t)(2 * n) * 15;
            const float* kp = Ke + (size_t)n * 8;
            float acc = 0.f;
#pragma unroll
            for (int m = 0; m < 8; ++m) acc += kp[m] * u[step * row[2 * m]];
            newW[2 * n] = oldW[n] + hf * acc;
        }
        // odd outputs: new[2n+1] = 0.5*(old[n]+old[n+1]) + hf * 15-tap gather
        for (int n = tid; n < Nc - 1; n += 256) {
            const int* row = nbr + (size_t)(2 * n + 1) * 15;
            const float* kp = Ko + (size_t)n * 15;
            float acc = 0.f;
#pragma unroll
            for (int m = 0; m < 15; ++m) acc += kp[m] * u[step * row[m]];
            newW[2 * n + 1] = 0.5f * (oldW[n] + oldW[n + 1]) + hf * acc;
        }
        __syncthreads();
        float* tmp = oldW; oldW = newW; newW = tmp;
    }

    // final store with boundary replacement from s @ Kbd_V^T * H
    for (int n = tid; n < 8193; n += 256) {
        float v = oldW[n];
        if (n == 0 || n == 8192) {
            const float* kv = Kbd_V + ((n == 0) ? 0 : 32);
            const float* sp = sbuf + bc * 32;
            float acc = 0.f;
#pragma unroll
            for (int r = 0; r < 32; ++r) acc += kv[r] * sp[r];
            v = H * acc;
        }
        out[base + n] = v;
    }
}

// ---------------------------------------------------------------------------
// Host launcher
// ---------------------------------------------------------------------------
extern "C" void kernel_launch(void* const* d_in, const int* in_sizes, int n_in,
                              void* d_out, int out_size, void* d_ws, size_t ws_size,
                              hipStream_t stream) {
    // setup_inputs() dict order:
    const float* uh     = (const float*)d_in[0];   // 512 x 8193
    const float* KHH_U  = (const float*)d_in[1];   // 1025 x 32
    const float* KHH_V  = (const float*)d_in[2];   // 1025 x 32
    const float* Ke0_U  = (const float*)d_in[3];   // 1025 x 32
    const float* Ke0_V  = (const float*)d_in[4];   // 8 x 32
    const float* Ko0_U  = (const float*)d_in[5];   // 1024 x 32
    const float* Ko0_V  = (const float*)d_in[6];   // 15 x 32
    const int*   nbr0   = (const int*)d_in[7];     // 8193 x 15
    const float* Ke1_U  = (const float*)d_in[8];   // 2049 x 32
    const float* Ke1_V  = (const float*)d_in[9];   // 8 x 32
    const float* Ko1_U  = (const float*)d_in[10];  // 2048 x 32
    const float* Ko1_V  = (const float*)d_in[11];  // 15 x 32
    const int*   nbr1   = (const int*)d_in[12];    // 4097 x 15
    const float* Ke2_U  = (const float*)d_in[13];  // 4097 x 32
    const float* Ke2_V  = (const float*)d_in[14];  // 8 x 32
    const float* Ko2_U  = (const float*)d_in[15];  // 4096 x 32
    const float* Ko2_V  = (const float*)d_in[16];  // 15 x 32
    const int*   nbr2   = (const int*)d_in[17];    // 2049 x 15
    const float* Kbd_U  = (const float*)d_in[18];  // 8193 x 32
    const float* Kbd_V  = (const float*)d_in[19];  // 2 x 32

    // workspace layout (floats)
    float* ws  = (float*)d_ws;
    float* wh0 = ws;                 // 512*1025 = 524800
    float* sb  = wh0 + 524800;       // 512*32   = 16384
    float* Ke0 = sb + 16384;         // 1025*8   = 8200
    float* Ke1 = Ke0 + 8200;         // 2049*8   = 16392
    float* Ke2 = Ke1 + 16392;        // 4097*8   = 32776
    float* Ko0 = Ke2 + 32776;        // 1024*15  = 15360
    float* Ko1 = Ko0 + 15360;        // 2048*15  = 30720
    float* Ko2 = Ko1 + 30720;        // 4096*15  = 61440

    float* outp = (float*)d_out;

    auto grid = [](int elems) { return (elems + 255) / 256; };

    // K0: materialize dense per-level kernels
    k_rankmm<<<grid(1025 * 8),  256, 0, stream>>>(Ke0_U, Ke0_V, Ke0, 1025, 8);
    k_rankmm<<<grid(1024 * 15), 256, 0, stream>>>(Ko0_U, Ko0_V, Ko0, 1024, 15);
    k_rankmm<<<grid(2049 * 8),  256, 0, stream>>>(Ke1_U, Ke1_V, Ke1, 2049, 8);
    k_rankmm<<<grid(2048 * 15), 256, 0, stream>>>(Ko1_U, Ko1_V, Ko1, 2048, 15);
    k_rankmm<<<grid(4097 * 8),  256, 0, stream>>>(Ke2_U, Ke2_V, Ke2, 4097, 8);
    k_rankmm<<<grid(4096 * 15), 256, 0, stream>>>(Ko2_U, Ko2_V, Ko2, 4096, 15);

    // K1: WMMA low-rank projections (t, s, wh0)
    k_wmma_proj<<<4, 256, 0, stream>>>(uh, KHH_U, KHH_V, Kbd_U, wh0, sb);

    // K2: fused 3-level refinement + boundary, one block per (b,c)
    k_refine<<<512, 256, 0, stream>>>(uh, wh0, sb, Kbd_V,
                                      Ke0, Ko0, Ke1, Ko1, Ke2, Ko2,
                                      nbr2, nbr1, nbr0, outp);
}